// SpectralMambaBlock_2705829396664
// MI455X (gfx1250) — compile-verified
//
#include <hip/hip_runtime.h>
#include <math.h>

// ---------------- constants of the problem ----------------
#define DMODEL 128
#define NF     65
#define F2     130     // 2*NF
#define DI     260     // EXPAND*F2
#define DTR    9       // (F2+15)/16
#define DBLW   41      // DTR + 2*16
#define DSTATE 16
#define SEQL   4096

typedef __attribute__((ext_vector_type(2))) float v2f;
typedef __attribute__((ext_vector_type(8))) float v8f;

// ---------------- device math helpers ----------------
__device__ __forceinline__ float siluf(float x) {
    return x / (1.f + expf(-x));
}
__device__ __forceinline__ float softplusf(float x) {
    return fmaxf(x, 0.f) + log1pf(expf(-fabsf(x)));
}
__device__ __forceinline__ float geluf(float x) {
    return 0.5f * x * (1.f + erff(x * 0.70710678118654752f));
}

enum { ACT_NONE = 0, ACT_SOFTPLUS = 1, ACT_GELU = 2, ACT_SILU = 3 };

// ---------------- DFT / IDFT weight generation ----------------
__global__ __launch_bounds__(256) void gen_dft_kernel(float* __restrict__ wfft,
                                                      float* __restrict__ widft) {
    const float inv  = 0.08838834764831845f;   // 1/sqrt(128)
    const float step = 6.283185307179586f / 128.f;
    int i = blockIdx.x * 256 + threadIdx.x;
    if (i < F2 * DMODEL) {
        int o = i / DMODEL, d = i - o * DMODEL;
        int k = (o < NF) ? o : (o - NF);
        float ang = (float)((k * d) & 127) * step;
        wfft[i] = (o < NF) ? cosf(ang) * inv : -sinf(ang) * inv;
    } else {
        int j2 = i - F2 * DMODEL;
        if (j2 < DMODEL * F2) {
            int d = j2 / F2, j = j2 - d * F2;
            float v;
            if (j == 0)                 v = inv;
            else if (j == 64)           v = (d & 1) ? -inv : inv;
            else if (j == 65 || j == 129) v = 0.f;
            else if (j < 64) {
                float ang = (float)((j * d) & 127) * step;
                v = 2.f * cosf(ang) * inv;
            } else {
                int k = j - NF;
                float ang = (float)((k * d) & 127) * step;
                v = -2.f * sinf(ang) * inv;
            }
            widft[j2] = v;
        }
    }
}

// ---------------- fragment loaders (branchless) ----------------
template <int STRIDE>
__device__ __forceinline__ v2f frag_load(const float* __restrict__ row, int k0) {
    v2f f;
    if constexpr ((STRIDE & 1) == 0) {
        // row base and k0 are 8-byte aligned for even strides
        f = *(const v2f*)(row + k0);
    } else {
        f.x = row[k0];
        f.y = row[k0 + 1];
    }
    return f;
}

// ---------------- WMMA f32 GEMM:  C = act( A @ W^T (+bias) ) ----------------
// A: M x K (row stride LDA), W: NC x K (row-major, stride K), C/R: M x NC.
// Out-of-range output columns accumulate garbage from W row 0 but are never
// stored, so the K main loop needs no guards at all.
template <int ACT, bool BIAS, bool RES, int NT, int LDA, int K, int NC>
__global__ __launch_bounds__(256) void gemm_wmma_f32(const float* __restrict__ A,
                                                     const float* __restrict__ W,
                                                     const float* __restrict__ bias,
                                                     const float* __restrict__ R,
                                                     float* __restrict__ C, int M) {
    constexpr int tiles_n  = (NC + 15) >> 4;
    constexpr int groups_n = (tiles_n + NT - 1) / NT;
    const int lane = threadIdx.x & 31;
    const int wave = threadIdx.x >> 5;
    const int tiles_m = M >> 4;
    int grp = blockIdx.x * 8 + wave;
    if (grp >= tiles_m * groups_n) return;          // wave-uniform exit
    const int tm = grp / groups_n;
    const int tg = grp - tm * groups_n;

    const int ln   = lane & 15;
    const int kb   = (lane >> 4) << 1;              // 0 or 2
    const int mrow = (tm << 4) + ln;

    const float* Arow = A + (size_t)mrow * LDA;
    int ncol[NT];
    bool nok[NT];
    const float* Wrow[NT];
#pragma unroll
    for (int i = 0; i < NT; ++i) {
        ncol[i] = ((tg * NT + i) << 4) + ln;
        nok[i]  = (ncol[i] < NC);
        Wrow[i] = W + (size_t)(nok[i] ? ncol[i] : 0) * K;
    }

    v8f acc[NT];
#pragma unroll
    for (int i = 0; i < NT; ++i) acc[i] = {0.f, 0.f, 0.f, 0.f, 0.f, 0.f, 0.f, 0.f};

    constexpr int KM = K & ~3;
#pragma unroll
    for (int k = 0; k < KM; k += 4) {
        const v2f a = frag_load<LDA>(Arow, k + kb);
#pragma unroll
        for (int i = 0; i < NT; ++i) {
            const v2f b = frag_load<K>(Wrow[i], k + kb);
            acc[i] = __builtin_amdgcn_wmma_f32_16x16x4_f32(false, a, false, b, (short)0,
                                                           acc[i], false, false);
        }
    }
    if constexpr (K & 3) {                          // tail: clamped loads + selects
        const int k0 = KM + kb, k1 = k0 + 1;
        const int c0 = (k0 < K) ? k0 : 0;
        const int c1 = (k1 < K) ? k1 : 0;
        v2f a;
        {
            const float ax = Arow[c0], ay = Arow[c1];
            a.x = (k0 < K) ? ax : 0.f;
            a.y = (k1 < K) ? ay : 0.f;
        }
#pragma unroll
        for (int i = 0; i < NT; ++i) {
            const float bx = Wrow[i][c0], by = Wrow[i][c1];
            v2f b;
            b.x = (k0 < K) ? bx : 0.f;
            b.y = (k1 < K) ? by : 0.f;
            acc[i] = __builtin_amdgcn_wmma_f32_16x16x4_f32(false, a, false, b, (short)0,
                                                           acc[i], false, false);
        }
    }

    const int rbase = (tm << 4) + ((lane >> 4) << 3);
#pragma unroll
    for (int i = 0; i < NT; ++i) {
        if (nok[i]) {
            const float bv = BIAS ? bias[ncol[i]] : 0.f;
#pragma unroll
            for (int r = 0; r < 8; ++r) {
                const int row = rbase + r;
                float v = acc[i][r] + bv;
                if constexpr (RES) v = R[(size_t)row * NC + ncol[i]] - v;
                if constexpr (ACT == ACT_SOFTPLUS) v = softplusf(v);
                else if constexpr (ACT == ACT_GELU) v = geluf(v);
                else if constexpr (ACT == ACT_SILU) v = siluf(v);
                C[(size_t)row * NC + ncol[i]] = v;
            }
        }
    }
}

// ---------------- depthwise causal conv4 + bias + silu ----------------
__global__ __launch_bounds__(256) void conv_silu_kernel(const float* __restrict__ xz,
                                                        const float* __restrict__ cw,
                                                        const float* __restrict__ cb,
                                                        float* __restrict__ uc, int L) {
    const int d = blockIdx.y * 256 + threadIdx.x;
    if (d >= DI) return;
    const int t = blockIdx.x;            // token index b*L + l
    const int b = t / L;
    const int l = t - b * L;
    float acc = cb[d];
    const float* base = xz + (size_t)b * L * (2 * DI) + d;
#pragma unroll
    for (int j = 0; j < 4; ++j) {
        const int ls = l - 3 + j;
        if (ls >= 0) acc += cw[d * 4 + j] * base[(size_t)ls * (2 * DI)];
    }
    uc[(size_t)t * DI + d] = siluf(acc);
}

// ---------------- selective scan + skip + silu(z) gate ----------------
__global__ __launch_bounds__(256) void scan_kernel(const float* __restrict__ dbl,
                                                   const float* __restrict__ dt,
                                                   const float* __restrict__ uc,
                                                   const float* __restrict__ xz,
                                                   const float* __restrict__ A_log,
                                                   const float* __restrict__ Dp,
                                                   float* __restrict__ y,
                                                   int chTotal, int L) {
    const int lane = threadIdx.x & 31;
    const int wave = threadIdx.x >> 5;
    const int s    = lane & 15;
    const int half = lane >> 4;
    const int ch   = (blockIdx.x * 8 + wave) * 2 + half;
    if (ch >= chTotal) return;
    const int b = ch / DI;
    const int d = ch - b * DI;

    const float Ads = -expf(A_log[d * DSTATE + s]);
    const float Dpd = Dp[d];
    float h = 0.f;

    const size_t tok0 = (size_t)b * L;
    for (int l = 0; l < L; ++l) {
        const size_t t = tok0 + l;
        const float dtv = dt[t * DI + d];
        const float uv  = uc[t * DI + d];
        const float Bv  = dbl[t * DBLW + DTR + s];
        const float Cv  = dbl[t * DBLW + DTR + DSTATE + s];
        const float dA  = expf(dtv * Ads);
        h = dA * h + (dtv * uv) * Bv;
        float p = h * Cv;
        p += __shfl_xor(p, 1);
        p += __shfl_xor(p, 2);
        p += __shfl_xor(p, 4);
        p += __shfl_xor(p, 8);
        if (s == 0) {
            const float zv = xz[t * (2 * DI) + DI + d];
            y[t * DI + d] = (p + uv * Dpd) * siluf(zv);
        }
    }
}

// ---------------- LayerNorm over 130 features (1 wave / token) ----------------
template <bool SUB, bool ADDRES>
__global__ __launch_bounds__(256) void ln_kernel(const float* __restrict__ X,
                                                 const float* __restrict__ S,
                                                 const float* __restrict__ g,
                                                 const float* __restrict__ bta,
                                                 const float* __restrict__ AR,
                                                 float* __restrict__ out, int M) {
    const int token = blockIdx.x * 8 + (threadIdx.x >> 5);
    if (token >= M) return;
    const int lane = threadIdx.x & 31;
    float vals[5];
    float sum = 0.f;
#pragma unroll
    for (int j = 0; j < 5; ++j) {
        const int i = lane + 32 * j;
        float v = 0.f;
        if (i < F2) {
            v = X[(size_t)token * F2 + i];
            if constexpr (SUB) v -= S[(size_t)token * F2 + i];
        }
        vals[j] = v;
        sum += v;
    }
    for (int off = 16; off; off >>= 1) sum += __shfl_xor(sum, off);
    const float mean = sum * (1.f / (float)F2);
    float var = 0.f;
#pragma unroll
    for (int j = 0; j < 5; ++j) {
        const int i = lane + 32 * j;
        const float dv = (i < F2) ? (vals[j] - mean) : 0.f;
        var += dv * dv;
    }
    for (int off = 16; off; off >>= 1) var += __shfl_xor(var, off);
    const float rs = rsqrtf(var * (1.f / (float)F2) + 1e-5f);
#pragma unroll
    for (int j = 0; j < 5; ++j) {
        const int i = lane + 32 * j;
        if (i < F2) {
            float v = (vals[j] - mean) * rs * g[i] + bta[i];
            if constexpr (ADDRES) v += AR[(size_t)token * F2 + i];
            out[(size_t)token * F2 + i] = v;
        }
    }
}

// ---------------- launch ----------------
template <int NT, int NC>
static inline int gemm_blocks(int M) {
    constexpr int tiles_n  = (NC + 15) >> 4;
    constexpr int groups_n = (tiles_n + NT - 1) / NT;
    const int t = (M >> 4) * groups_n;
    return (t + 7) / 8;
}

extern "C" void kernel_launch(void* const* d_in, const int* in_sizes, int n_in,
                              void* d_out, int out_size, void* d_ws, size_t ws_size,
                              hipStream_t stream) {
    const float* x         = (const float*)d_in[0];
    const float* in_proj_w = (const float*)d_in[1];
    const float* conv_w    = (const float*)d_in[2];
    const float* conv_b    = (const float*)d_in[3];
    const float* x_proj_w  = (const float*)d_in[4];
    const float* dt_proj_w = (const float*)d_in[5];
    const float* dt_proj_b = (const float*)d_in[6];
    const float* A_log     = (const float*)d_in[7];
    const float* Dp        = (const float*)d_in[8];
    const float* mamba_w   = (const float*)d_in[9];
    const float* ln1_g     = (const float*)d_in[10];
    const float* ln1_b     = (const float*)d_in[11];
    const float* ff_w1     = (const float*)d_in[12];
    const float* ff_b1     = (const float*)d_in[13];
    const float* ff_w2     = (const float*)d_in[14];
    const float* ff_b2     = (const float*)d_in[15];
    const float* ln2_g     = (const float*)d_in[16];
    const float* ln2_b     = (const float*)d_in[17];

    const int M  = in_sizes[0] / DMODEL;   // 32768 tokens
    const int Bn = M / SEQL;               // 8

    // workspace layout (floats); all offsets stay 16B-aligned
    float* wfft  = (float*)d_ws;                 // 130*128
    float* widft = wfft + F2 * DMODEL;           // 128*130
    float* xf    = widft + DMODEL * F2;          // M*130
    float* xz    = xf + (size_t)M * F2;          // M*520
    float* uc    = xz + (size_t)M * 2 * DI;      // M*260
    float* dbl   = uc + (size_t)M * DI;          // M*41
    float* dtb   = dbl + (size_t)M * DBLW;       // M*260  (dt; y aliases it)
    float* mbuf  = dtb + (size_t)M * DI;         // M*130  (m; later spec)
    float* r1n   = mbuf + (size_t)M * F2;        // M*130
    float* ybuf  = dtb;                          // alias: read-then-write per (t,d)
    float* h1    = xz;                           // alias: xz dead after scan
    float* pre   = uc;                           // alias: uc dead after scan
    float* spec  = mbuf;                         // alias: m dead after ln1

    // 0) DFT / IDFT weights
    gen_dft_kernel<<<(2 * F2 * DMODEL + 255) / 256, 256, 0, stream>>>(wfft, widft);

    // 1) rFFT as GEMM: xf = x @ Wfft^T        (M x 128) @ (128 x 130)
    gemm_wmma_f32<ACT_NONE, false, false, 3, DMODEL, DMODEL, F2>
        <<<gemm_blocks<3, F2>(M), 256, 0, stream>>>(x, wfft, nullptr, nullptr, xf, M);

    // 2) in_proj: xz = xf @ in_proj_w^T       (M x 130) @ (130 x 520)
    gemm_wmma_f32<ACT_NONE, false, false, 3, F2, F2, 2 * DI>
        <<<gemm_blocks<3, 2 * DI>(M), 256, 0, stream>>>(xf, in_proj_w, nullptr, nullptr,
                                                        xz, M);

    // 3) depthwise causal conv + silu: uc
    conv_silu_kernel<<<dim3(M, 2), 256, 0, stream>>>(xz, conv_w, conv_b, uc, SEQL);

    // 4) x_proj: dbl = uc @ x_proj_w^T        (M x 260) @ (260 x 41)
    gemm_wmma_f32<ACT_NONE, false, false, 3, DI, DI, DBLW>
        <<<gemm_blocks<3, DBLW>(M), 256, 0, stream>>>(uc, x_proj_w, nullptr, nullptr,
                                                      dbl, M);

    // 5) dt = softplus(dbl[:, :9] @ dt_proj_w^T + dt_proj_b)
    gemm_wmma_f32<ACT_SOFTPLUS, true, false, 3, DBLW, DTR, DI>
        <<<gemm_blocks<3, DI>(M), 256, 0, stream>>>(dbl, dt_proj_w, dt_proj_b, nullptr,
                                                    dtb, M);

    // 6) selective scan + skip + silu(z) gate -> y
    {
        const int chTotal = Bn * DI;               // 2080
        const int waves   = (chTotal + 1) / 2;
        const int blocks  = (waves + 7) / 8;
        scan_kernel<<<blocks, 256, 0, stream>>>(dbl, dtb, uc, xz, A_log, Dp, ybuf,
                                                chTotal, SEQL);
    }

    // 7) mamba out: m = y @ mamba_out_w^T     (M x 260) @ (260 x 130)
    gemm_wmma_f32<ACT_NONE, false, false, 3, DI, DI, F2>
        <<<gemm_blocks<3, F2>(M), 256, 0, stream>>>(ybuf, mamba_w, nullptr, nullptr,
                                                    mbuf, M);

    // 8) r1n = LN(xf - m)
    ln_kernel<true, false><<<(M + 7) / 8, 256, 0, stream>>>(xf, mbuf, ln1_g, ln1_b,
                                                            nullptr, r1n, M);

    // 9) h1 = gelu(r1n @ ff_w1^T + b1)        (M x 130) @ (130 x 260)
    gemm_wmma_f32<ACT_GELU, true, false, 3, F2, F2, DI>
        <<<gemm_blocks<3, DI>(M), 256, 0, stream>>>(r1n, ff_w1, ff_b1, nullptr, h1, M);

    // 10) pre = silu(r1n - (h1 @ ff_w2^T + b2))
    gemm_wmma_f32<ACT_SILU, true, true, 3, DI, DI, F2>
        <<<gemm_blocks<3, F2>(M), 256, 0, stream>>>(h1, ff_w2, ff_b2, r1n, pre, M);

    // 11) spec = LN(pre)*g+b + xf
    ln_kernel<false, true><<<(M + 7) / 8, 256, 0, stream>>>(pre, nullptr, ln2_g, ln2_b,
                                                            xf, spec, M);

    // 12) irFFT as GEMM: out = spec @ Widft^T (M x 130) @ (130 x 128)
    gemm_wmma_f32<ACT_NONE, false, false, 4, F2, F2, DMODEL>
        <<<gemm_blocks<4, DMODEL>(M), 256, 0, stream>>>(spec, widft, nullptr, nullptr,
                                                        (float*)d_out, M);
}